// PostProcess_19026705121683
// MI455X (gfx1250) — compile-verified
//
#include <hip/hip_runtime.h>
#include <stdint.h>

// ---------------- tunables / constants ----------------
#define PT          128      // priors per block in scoring kernel
#define K1_THREADS  256      // 8 waves (wave32)
#define TOPK        1000     // PRE_NMS_TOPK
#define OUTK        100      // DET_PER_IMG
#define SELBUF      4096     // per-image collected-candidate buffer (pow2, bitonic)
#define NBINS       2048     // score histogram bins
#define NW          32       // 32x32 = 1024 bits >= TOPK keep/suppress words
#define SCORE_TH    0.05f
#define NMS_TH      0.45f
#define MIN_SZ      0.01f

typedef __attribute__((ext_vector_type(2))) float v2f;
typedef __attribute__((ext_vector_type(8))) float v8f;

// ---------------- CDNA5 async global->LDS helpers ----------------
typedef __attribute__((address_space(1))) int g1int;   // global
typedef __attribute__((address_space(3))) int l3int;   // LDS

__device__ __forceinline__ unsigned lds_off(const void* p) {
  // generic "shared" pointer low 32 bits == LDS byte offset (aperture scheme)
  return (unsigned)(unsigned long long)p;
}

__device__ __forceinline__ void async_g2l_b32(const float* g, float* l) {
#if __has_builtin(__builtin_amdgcn_global_load_async_to_lds_b32)
  __builtin_amdgcn_global_load_async_to_lds_b32(
      (g1int*)(unsigned long long)g,
      (l3int*)(unsigned long long)lds_off(l),
      0, 0);
#else
  asm volatile("global_load_async_to_lds_b32 %0, %1, off"
               :: "v"(lds_off(l)), "v"((unsigned long long)g)
               : "memory");
#endif
}

__device__ __forceinline__ void wait_async0() {
#if __has_builtin(__builtin_amdgcn_s_wait_asynccnt)
  __builtin_amdgcn_s_wait_asynccnt(0);
#else
  asm volatile("s_wait_asynccnt 0" ::: "memory");
#endif
}

// ---------------- kernel 0: init counters ----------------
__global__ void k_init(int* candCount, int B) {
  int t = blockIdx.x * blockDim.x + threadIdx.x;
  if (t < B) candCount[t] = 0;
}

// ---------------- kernel 1: softmax + decode + threshold append ----------------
// grid = B * bpi blocks of 256 threads; each wave owns 16 priors.
__global__ void k_score(const float* __restrict__ logits,
                        const float* __restrict__ relbox,
                        const float* __restrict__ priors,
                        const float* __restrict__ tsz,
                        float* __restrict__ boxesS,
                        int* __restrict__ candCount,
                        unsigned long long* __restrict__ keys,
                        int B, int P, int C, int bpi, int CAPr) {
  extern __shared__ float smem[];
  float* tile = smem;                     // PT * C logits
  float* wsum = smem + PT * C;            // 8 waves * 16 row sums

  const int t    = threadIdx.x;
  const int lane = t & 31;
  const int wave = t >> 5;
  const int img  = blockIdx.x / bpi;
  const int base = (blockIdx.x % bpi) * PT;

  int hi = base + PT; if (hi > P) hi = P;
  const int nDw = (hi - base) * C;

  // fill invalid tail of tile (harmless rows; results never stored)
  for (int i = t; i < PT * C; i += K1_THREADS)
    if (i >= nDw) tile[i] = -1e30f;

  // async-stage the logits tile into LDS (ASYNCcnt path)
  const float* src = logits + ((size_t)img * P + base) * (size_t)C;
  for (int i = t; i < nDw; i += K1_THREADS)
    async_g2l_b32(src + i, tile + i);

  const int  row     = lane & 15;
  const int  half    = lane >> 4;        // 0: K=0,1 ; 1: K=2,3 (A-matrix layout)
  const int  tileRow = wave * 16 + row;
  const int  prior   = base + tileRow;
  const bool pvalid  = prior < P;
  const int  pc      = pvalid ? prior : (P - 1);

  // ---- box decode (variances 0.1 / 0.2); overlaps the async tile fill ----
  const float scW = tsz[img * 2 + 1];
  const float scH = tsz[img * 2 + 0];
  const float pcx = priors[pc * 4 + 0], pcy = priors[pc * 4 + 1];
  const float pw  = priors[pc * 4 + 2], ph  = priors[pc * 4 + 3];
  const float* rb = relbox + ((size_t)img * P + pc) * 4;
  const float cx = pcx + rb[0] * 0.1f * pw;
  const float cy = pcy + rb[1] * 0.1f * ph;
  const float bw = pw * __expf(rb[2] * 0.2f);
  const float bh = ph * __expf(rb[3] * 0.2f);
  const float x1 = (cx - 0.5f * bw) * scW, y1 = (cy - 0.5f * bh) * scH;
  const float x2 = (cx + 0.5f * bw) * scW, y2 = (cy + 0.5f * bh) * scH;
  const bool size_ok = ((x2 - x1) >= MIN_SZ) && ((y2 - y1) >= MIN_SZ);
  if (pvalid && half == 0) {
    float* o = boxesS + ((size_t)img * P + prior) * 4;
    o[0] = x1; o[1] = y1; o[2] = x2; o[3] = y2;
  }

  wait_async0();
  __syncthreads();

  // ---- row max for stable softmax ----
  const float* trow = tile + (size_t)tileRow * C;
  float rmax = -1e30f;
  for (int k = 0; k < C; ++k) rmax = fmaxf(rmax, trow[k]);

  // ---- denominators via WMMA f32 16x16x4 row-sum (B = ones) ----
  // A layout (32-bit 16x4): lanes 0-15 -> K=0(v0),K=1(v1); lanes 16-31 -> K=2,K=3.
  v8f acc = {};
  v2f bones; bones.x = 1.0f; bones.y = 1.0f;
  const int fullChunks = C >> 2;          // branch-free straight-line chunks
  for (int c = 0; c < fullChunks; ++c) {
    const int k0 = 4 * c + 2 * half;
    v2f a;
    a.x = __expf(trow[k0]     - rmax);
    a.y = __expf(trow[k0 + 1] - rmax);
    acc = __builtin_amdgcn_wmma_f32_16x16x4_f32(false, a, false, bones,
                                                (short)0, acc, false, false);
  }
  if (C & 3) {                            // tail chunk: clamped load + cndmask
    const int k0 = (fullChunks << 2) + 2 * half;
    const int k0c = (k0 < C - 1) ? k0 : (C - 1);
    const float e0 = __expf(trow[k0c]                           - rmax);
    const float e1 = __expf(trow[(k0c + 1 < C) ? k0c + 1 : k0c] - rmax);
    v2f a;
    a.x = (k0     < C) ? e0 : 0.0f;
    a.y = (k0 + 1 < C) ? e1 : 0.0f;
    acc = __builtin_amdgcn_wmma_f32_16x16x4_f32(false, a, false, bones,
                                                (short)0, acc, false, false);
  }
  // C/D layout: VGPR r, lanes 0-15 hold M=r (col=lane), lanes 16-31 hold M=8+r.
  if (lane == 0 || lane == 16) {
    const int mb = (lane == 0) ? 0 : 8;
#pragma unroll
    for (int r = 0; r < 8; ++r) wsum[wave * 16 + mb + r] = acc[r];
  }
  __syncthreads();
  const float inv = 1.0f / wsum[wave * 16 + row];

  // ---- threshold scan: two lanes per row split the C-1 foreground classes ----
  if (pvalid && size_ok) {
    for (int c0 = half; c0 < C - 1; c0 += 2) {
      const float sc = __expf(trow[c0 + 1] - rmax) * inv;
      if (sc > SCORE_TH) {
        const unsigned flat = (unsigned)(prior * (C - 1) + c0);
        const unsigned long long key =
            ((unsigned long long)__float_as_uint(sc) << 32) |
            (unsigned long long)(0xFFFFFFFFu - flat);   // score desc, index asc
        const int pos = atomicAdd(&candCount[img], 1);
        if (pos < CAPr) keys[(size_t)img * CAPr + pos] = key;
      }
    }
  }
}

// ---------------- kernel 2: per-image exact top-1000 selection ----------------
__global__ void k_select(const unsigned long long* __restrict__ keys,
                         const int* __restrict__ candCount,
                         float* __restrict__ tS, int* __restrict__ tPr,
                         int* __restrict__ tCl, int* __restrict__ tVa,
                         int C, int CAPr) {
  __shared__ int hist[NBINS];
  __shared__ unsigned long long buf[SELBUF];
  __shared__ int nsel;
  __shared__ int cutbin;

  const int img = blockIdx.x, t = threadIdx.x, NT = blockDim.x;
  const unsigned long long* kk = keys + (size_t)img * CAPr;
  int n = candCount[img]; if (n > CAPr) n = CAPr;

  for (int i = t; i < NBINS; i += NT) hist[i] = 0;
  if (t == 0) nsel = 0;
  __syncthreads();

  for (int i = t; i < n; i += NT) {
    const unsigned sb = (unsigned)(kk[i] >> 32);
    atomicAdd(&hist[(sb >> 15) & (NBINS - 1)], 1);
  }
  __syncthreads();

  if (t == 0) {
    int acc = 0, cb = 0;
    for (int b = NBINS - 1; b >= 0; --b) {
      acc += hist[b];
      if (acc >= TOPK) { cb = b; break; }
    }
    cutbin = (acc >= TOPK) ? cb : 0;
  }
  __syncthreads();

  const int cb = cutbin;
  for (int i = t; i < n; i += NT) {
    const unsigned long long key = kk[i];
    const unsigned sb = (unsigned)(key >> 32);
    if ((int)((sb >> 15) & (NBINS - 1)) >= cb) {
      const int p = atomicAdd(&nsel, 1);
      if (p < SELBUF) buf[p] = key;
    }
  }
  __syncthreads();
  int m = nsel; if (m > SELBUF) m = SELBUF;
  for (int i = t; i < SELBUF; i += NT)
    if (i >= m) buf[i] = 0ull;
  __syncthreads();

  // bitonic sort ascending (64-bit keys, DS traffic)
  for (int k = 2; k <= SELBUF; k <<= 1) {
    for (int j = k >> 1; j > 0; j >>= 1) {
      for (int e = t; e < SELBUF; e += NT) {
        const int ixj = e ^ j;
        if (ixj > e) {
          const unsigned long long a = buf[e], b2 = buf[ixj];
          const bool up = ((e & k) == 0);
          if (up ? (a > b2) : (a < b2)) { buf[e] = b2; buf[ixj] = a; }
        }
      }
      __syncthreads();
    }
  }

  for (int r = t; r < 1024; r += NT) {
    const unsigned long long key = buf[SELBUF - 1 - r];
    const float sc = __uint_as_float((unsigned)(key >> 32));
    const unsigned flat = 0xFFFFFFFFu - (unsigned)(key & 0xFFFFFFFFu);
    const int pr = (key != 0ull) ? (int)(flat / (unsigned)(C - 1)) : 0;
    const int c0 = (key != 0ull) ? (int)(flat % (unsigned)(C - 1)) : 0;
    const size_t o = (size_t)img * 1024 + r;
    tS[o] = sc; tPr[o] = pr; tCl[o] = c0;
    tVa[o] = (r < TOPK) && (sc > SCORE_TH);
  }
}

// ---------------- kernel 3: bitmask NMS + finalize top-100 ------------------
__device__ __forceinline__ float iou4(const float* a, const float* b) {
  const float areaA = (a[2] - a[0]) * (a[3] - a[1]);
  const float areaB = (b[2] - b[0]) * (b[3] - b[1]);
  const float lx = fmaxf(a[0], b[0]), ly = fmaxf(a[1], b[1]);
  const float rx = fminf(a[2], b[2]), ry = fminf(a[3], b[3]);
  const float w = fmaxf(rx - lx, 0.0f), h = fmaxf(ry - ly, 0.0f);
  const float inter = w * h;
  return inter / (areaA + areaB - inter + 1e-12f);
}

__global__ void k_nms(const float* __restrict__ boxesS,
                      const float* __restrict__ tS, const int* __restrict__ tPr,
                      const int* __restrict__ tCl, const int* __restrict__ tVa,
                      float* __restrict__ out, int B, int P) {
  extern __shared__ unsigned sup[];  // [TOPK][NW] suppression bit-matrix, 128000 B
  __shared__ float bx[TOPK * 4];     // class-offset boxes
  __shared__ float ob[TOPK * 4];     // original scaled boxes
  __shared__ unsigned keepW[NW];
  __shared__ int maxcBits;

  const int img = blockIdx.x, t = threadIdx.x, NT = blockDim.x;
  const size_t tb = (size_t)img * 1024;

  if (t == 0) maxcBits = 0;          // float 0.0 bits; valid coords >= 0 dominate
  if (t < NW) keepW[t] = 0u;
  __syncthreads();

  float b0 = 0, b1 = 0, b2 = 0, b3 = 0;
  int va = 0, c0 = 0;
  if (t < TOPK) {
    va = tVa[tb + t];
    c0 = tCl[tb + t];
    const int pr = va ? tPr[tb + t] : 0;
    const float* bp = boxesS + ((size_t)img * P + pr) * 4;
    if (va) { b0 = bp[0]; b1 = bp[1]; b2 = bp[2]; b3 = bp[3]; }
    ob[t * 4 + 0] = b0; ob[t * 4 + 1] = b1;
    ob[t * 4 + 2] = b2; ob[t * 4 + 3] = b3;
    if (va) {
      atomicOr(&keepW[t >> 5], 1u << (t & 31));
      const float mx = fmaxf(fmaxf(b0, b1), fmaxf(b2, b3));
      if (mx > 0.0f) atomicMax(&maxcBits, __float_as_int(mx));
    }
  }
  __syncthreads();

  if (t < TOPK) {
    const float maxc = __int_as_float(maxcBits);
    const float off = (float)c0 * (maxc + 1.0f);
    bx[t * 4 + 0] = b0 + off; bx[t * 4 + 1] = b1 + off;
    bx[t * 4 + 2] = b2 + off; bx[t * 4 + 3] = b3 + off;
  }
  __syncthreads();

  // ---- fully parallel suppression matrix: word (i,w) owned by one thread ----
  for (int idx = t; idx < TOPK * NW; idx += NT) {
    const int i = idx >> 5;          // NW == 32
    const int w = idx & 31;
    float bi[4];
    bi[0] = bx[i * 4 + 0]; bi[1] = bx[i * 4 + 1];
    bi[2] = bx[i * 4 + 2]; bi[3] = bx[i * 4 + 3];
    unsigned mword = 0;
    const int jbase = w << 5;
    for (int jj = 0; jj < 32; ++jj) {
      const int j = jbase + jj;
      if (j < TOPK && j != i) {
        if (iou4(bi, &bx[j * 4]) > NMS_TH) mword |= (1u << jj);
      }
    }
    sup[idx] = mword;
  }
  __syncthreads();

  // ---- greedy pass on a single wave32: no block barriers, shfl broadcast ----
  if (t < 32) {
    unsigned kw = keepW[t];                       // lane t owns 32 keep bits
    for (int i = 0; i < TOPK; ++i) {
      const unsigned srcw = (unsigned)__shfl((int)kw, i >> 5, 32);
      if ((srcw >> (i & 31)) & 1u) {              // uniform branch across wave
        kw &= ~sup[i * NW + t];                   // conflict-free ds_load
        if (t == (i >> 5)) kw |= (1u << (i & 31));// self-bit never suppressed
      }
    }
    keepW[t] = kw;
  }
  __syncthreads();

  // ---- finalize: first (<=100) kept in score order ----
  if (t == 0) {
    float* oB = out;
    float* oS = out + (size_t)B * OUTK * 4;
    float* oL = oS + (size_t)B * OUTK;
    float* oV = oL + (size_t)B * OUTK;
    int m = 0;
    for (int i = 0; i < TOPK && m < OUTK; ++i) {
      if ((keepW[i >> 5] >> (i & 31)) & 1u) {
        const size_t o = (size_t)img * OUTK + m;
        oB[o * 4 + 0] = ob[i * 4 + 0]; oB[o * 4 + 1] = ob[i * 4 + 1];
        oB[o * 4 + 2] = ob[i * 4 + 2]; oB[o * 4 + 3] = ob[i * 4 + 3];
        oS[o] = tS[tb + i];
        oL[o] = (float)(tCl[tb + i] + 1);
        oV[o] = 1.0f;
        ++m;
      }
    }
    for (; m < OUTK; ++m) {
      const size_t o = (size_t)img * OUTK + m;
      oB[o * 4 + 0] = 0.0f; oB[o * 4 + 1] = 0.0f;
      oB[o * 4 + 2] = 0.0f; oB[o * 4 + 3] = 0.0f;
      oS[o] = 0.0f; oL[o] = -1.0f; oV[o] = 0.0f;
    }
  }
}

// ---------------- host launcher ----------------
extern "C" void kernel_launch(void* const* d_in, const int* in_sizes, int n_in,
                              void* d_out, int out_size, void* d_ws, size_t ws_size,
                              hipStream_t stream) {
  const float* logits = (const float*)d_in[0];   // [B,P,C]
  const float* relbox = (const float*)d_in[1];   // [B,P,4]
  const float* priors = (const float*)d_in[2];   // [P,4]
  const float* tsz    = (const float*)d_in[3];   // [B,2]

  const int B = in_sizes[3] / 2;
  const int P = in_sizes[2] / 4;
  const int C = (int)((long long)in_sizes[0] / ((long long)B * (long long)P));
  const int bpi = (P + PT - 1) / PT;

  // workspace carve
  char* w = (char*)d_ws;
  size_t off = 0;
  float* boxesS = (float*)(w + off); off += (size_t)B * P * 4 * sizeof(float);
  int* candCount = (int*)(w + off);  off += 256;
  float* tS = (float*)(w + off);     off += (size_t)B * 1024 * sizeof(float);
  int* tPr = (int*)(w + off);        off += (size_t)B * 1024 * sizeof(int);
  int* tCl = (int*)(w + off);        off += (size_t)B * 1024 * sizeof(int);
  int* tVa = (int*)(w + off);        off += (size_t)B * 1024 * sizeof(int);
  off = (off + 7) & ~(size_t)7;
  unsigned long long* keys = (unsigned long long*)(w + off);
  const size_t rem = (ws_size > off) ? (ws_size - off) : 0;
  long long capL = (long long)(rem / (8ull * (size_t)(B > 0 ? B : 1)));
  int CAPr = (capL > 131072) ? 131072 : (int)capL;
  if (CAPr < 4096) CAPr = 4096;   // minimum viable candidate budget

  k_init<<<dim3((B + 63) / 64), dim3(64), 0, stream>>>(candCount, B);

  const size_t lds1 = ((size_t)PT * C + 8 * 16) * sizeof(float);
  k_score<<<dim3(B * bpi), dim3(K1_THREADS), lds1, stream>>>(
      logits, relbox, priors, tsz, boxesS, candCount, keys, B, P, C, bpi, CAPr);

  k_select<<<dim3(B), dim3(1024), 0, stream>>>(
      keys, candCount, tS, tPr, tCl, tVa, C, CAPr);

  const size_t ldsN = (size_t)TOPK * NW * sizeof(unsigned);
  k_nms<<<dim3(B), dim3(1024), ldsN, stream>>>(
      boxesS, tS, tPr, tCl, tVa, (float*)d_out, B, P);
}